// cGRU_38431367365138
// MI455X (gfx1250) — compile-verified
//
#include <hip/hip_runtime.h>
#include <math.h>

#define B_ 64
#define T_ 1024
#define F_ 256
#define H_ 256
#define PLANE_ (F_ * H_)  // 65536 floats per packed weight plane

typedef __attribute__((ext_vector_type(2))) float v2f;
typedef __attribute__((ext_vector_type(8))) float v8f;

static __device__ __forceinline__ v8f wmma4(v2f a, v2f b, v8f c) {
  // D = A(16x4 f32) * B(4x16 f32) + C(16x16 f32)
  return __builtin_amdgcn_wmma_f32_16x16x4_f32(false, a, false, b, (short)0, c,
                                               false, false);
}

static __device__ __forceinline__ float fast_sigmoid(float x) {
  return 1.f / (1.f + __expf(-x));
}
static __device__ __forceinline__ float fast_tanh(float x) {
  // tanh(x) = 1 - 2/(exp(2x)+1); exact limits at +/-inf, no branches
  return 1.f - 2.f / (__expf(2.f * x) + 1.f);
}

struct Acc { v8f re; v8f im; };

// Complex 16x16 tile GEMM with fully pre-separated operands (no repack VALU)
// and two independent accumulator pairs (even/odd k-chunks) to halve the
// dependent-WMMA chain depth.
//  A: planar re/im planes, row stride aStride floats; lane loads b64
//     {A[row, k0], A[row, k0+1]} with k0 = kk + 2*(lane>=16).
//  B: packed planes: float2 {W[k0,col], W[k0+1,col]} at (k0/2)*H + col.
//     bN plane holds -im so the complex product needs no negation.
static __device__ __forceinline__ void cgemm_packed(
    const float* __restrict__ aRe, const float* __restrict__ aIm, int aStride,
    int row0, const float* __restrict__ bRe, const float* __restrict__ bIm,
    const float* __restrict__ bN, int col0, int K, Acc& acc) {
  const int lane = threadIdx.x & 31;
  const int mrow = lane & 15;
  const int half = lane >> 4;
  const float* ar = aRe + (size_t)(row0 + mrow) * aStride + 2 * half;
  const float* ai = aIm + (size_t)(row0 + mrow) * aStride + 2 * half;
  const size_t bo = ((size_t)half * H_ + col0 + mrow) * 2;
  const float* pr = bRe + bo;
  const float* pi = bIm + bo;
  const float* pn = bN + bo;

  const v8f z8 = {0.f, 0.f, 0.f, 0.f, 0.f, 0.f, 0.f, 0.f};
  v8f re0 = z8, im0 = z8, re1 = z8, im1 = z8;

#pragma unroll 2
  for (int kk = 0; kk < K; kk += 8) {
    // even chunk -> accumulator pair 0
    v2f aR0 = *(const v2f*)ar;
    v2f aI0 = *(const v2f*)ai;
    v2f bR0 = *(const v2f*)pr;
    v2f bI0 = *(const v2f*)pi;
    v2f bM0 = *(const v2f*)pn;
    // odd chunk -> accumulator pair 1
    v2f aR1 = *(const v2f*)(ar + 4);
    v2f aI1 = *(const v2f*)(ai + 4);
    v2f bR1 = *(const v2f*)(pr + 4 * H_);
    v2f bI1 = *(const v2f*)(pi + 4 * H_);
    v2f bM1 = *(const v2f*)(pn + 4 * H_);
    ar += 8; ai += 8;
    pr += 8 * H_; pi += 8 * H_; pn += 8 * H_;

    re0 = wmma4(aR0, bR0, re0);  // +ar*br
    im0 = wmma4(aR0, bI0, im0);  // +ar*bi
    re1 = wmma4(aR1, bR1, re1);
    im1 = wmma4(aR1, bI1, im1);
    re0 = wmma4(aI0, bM0, re0);  // -ai*bi (pre-negated plane)
    im0 = wmma4(aI0, bR0, im0);  // +ai*br
    re1 = wmma4(aI1, bM1, re1);
    im1 = wmma4(aI1, bR1, im1);
  }
#pragma unroll
  for (int q = 0; q < 8; ++q) {
    acc.re[q] = re0[q] + re1[q];
    acc.im[q] = im0[q] + im1[q];
  }
}

// ------- Kernel 0: pack weights into planar fragment layout; split c -------
// wpk: 6 matrices x {re, im, -im} planes of F_*H_ floats.
__global__ __launch_bounds__(256) void cgru_pack_kernel(
    const float* __restrict__ Wir, const float* __restrict__ Wiz,
    const float* __restrict__ Win, const float* __restrict__ Whr,
    const float* __restrict__ Whz, const float* __restrict__ Whn,
    const float* __restrict__ c, float* __restrict__ wpk,
    float* __restrict__ h0re, float* __restrict__ h0im) {
  const int idx = blockIdx.x * 256 + threadIdx.x;
  const int total = 6 * (F_ / 2) * H_;  // 196608 float2 sites
  if (idx < total) {
    const int w = idx / ((F_ / 2) * H_);
    const int rem = idx % ((F_ / 2) * H_);
    const int kp = rem / H_;   // k-pair
    const int col = rem % H_;
    const float* src = (w == 0) ? Wir : (w == 1) ? Wiz : (w == 2) ? Win
                     : (w == 3) ? Whr : (w == 4) ? Whz : Whn;
    float2 e0 = *(const float2*)(src + ((size_t)(2 * kp) * H_ + col) * 2);
    float2 e1 = *(const float2*)(src + ((size_t)(2 * kp + 1) * H_ + col) * 2);
    float* base = wpk + (size_t)w * 3 * PLANE_;
    size_t o = ((size_t)kp * H_ + col) * 2;
    *(float2*)(base + o) = make_float2(e0.x, e1.x);
    *(float2*)(base + PLANE_ + o) = make_float2(e0.y, e1.y);
    *(float2*)(base + 2 * PLANE_ + o) = make_float2(-e0.y, -e1.y);
  } else {
    const int i = idx - total;
    if (i < B_ * H_) {
      float2 cv = *(const float2*)(c + (size_t)i * 2);
      h0re[i] = cv.x;
      h0im[i] = cv.y;
    }
  }
}

// ---------------- Kernel 1: parallel input projections -----------------
// One WG per 16-row tile of xs; deinterleave tile to LDS planes once, then
// 8 waves x 6 (colTile,gate) tiles. Outputs interleaved complex.
__global__ __launch_bounds__(256) void cgru_proj_kernel(
    const float* __restrict__ xs, const float* __restrict__ wpk,
    const float* __restrict__ bir, const float* __restrict__ biz,
    const float* __restrict__ bin_, float* __restrict__ xr,
    float* __restrict__ xz, float* __restrict__ ysxn) {
  __shared__ float sre[16][F_ + 2];  // +2 pad: conflict-free b64 rows
  __shared__ float sim[16][F_ + 2];
  const int row0 = blockIdx.x * 16;

  // cooperative deinterleave: 16 rows x 128 complex-pairs
  for (int i = threadIdx.x; i < 16 * (F_ / 2); i += 256) {
    int r = i >> 7;
    int cp = i & 127;
    float4 v = *(const float4*)(xs + ((size_t)(row0 + r) * F_ + 2 * cp) * 2);
    *(float2*)&sre[r][2 * cp] = make_float2(v.x, v.z);
    *(float2*)&sim[r][2 * cp] = make_float2(v.y, v.w);
  }
  __syncthreads();

  const int wv = threadIdx.x >> 5;
  const int lane = threadIdx.x & 31;
  const int n = lane & 15, half = lane >> 4;

  for (int j = 0; j < 6; ++j) {
    const int cg = wv * 6 + j;  // 48 = 3 gates x 16 colTiles
    const int gate = cg >> 4;
    const int col0 = (cg & 15) * 16;
    const float* base = wpk + (size_t)gate * 3 * PLANE_;
    const float* bias = (gate == 0) ? bir : (gate == 1) ? biz : bin_;
    float* out = (gate == 0) ? xr : (gate == 1) ? xz : ysxn;

    Acc acc;
    cgemm_packed(&sre[0][0], &sim[0][0], F_ + 2, 0, base, base + PLANE_,
                 base + 2 * PLANE_, col0, F_, acc);

    float2 bb = *(const float2*)(bias + (size_t)(col0 + n) * 2);
#pragma unroll
    for (int q = 0; q < 8; ++q) {
      int m = q + 8 * half;
      *(float2*)(out + ((size_t)(row0 + m) * H_ + col0 + n) * 2) =
          make_float2(acc.re[q] + bb.x, acc.im[q] + bb.y);
    }
  }
}

// ---------------- Kernel 2: persistent cooperative scan -----------------
// 64 WGs: (btile 0..3) x (colTile 0..15); 3 waves/WG = gates r,z,n.
// h kept planar (re/im planes) so A-fragments are clean b64 loads.
__global__ __launch_bounds__(96) void cgru_scan_kernel(
    const float* __restrict__ xr, const float* __restrict__ xz,
    float* __restrict__ ys,  // holds xn on entry; becomes h_t
    const float* __restrict__ wpk, const float* __restrict__ bhn,
    float* __restrict__ h0re, float* __restrict__ h0im,
    float* __restrict__ h1re, float* __restrict__ h1im,
    unsigned* __restrict__ ctr, float* __restrict__ carry) {
  const int btile = blockIdx.x & 3;
  const int colTile = blockIdx.x >> 2;
  const int gate = threadIdx.x >> 5;
  const int lane = threadIdx.x & 31;
  const int n = lane & 15, half = lane >> 4;
  const int row0 = btile * 16, col0 = colTile * 16;

  const float* base = wpk + (size_t)(3 + gate) * 3 * PLANE_;  // Whr/Whz/Whn
  const float* xg = (gate == 0) ? xr : (gate == 1) ? xz : ys;

  __shared__ float rz[2][16][16][2];  // r and z tiles (complex)
  float* hre[2] = {h0re, h1re};
  float* him[2] = {h0im, h1im};

  for (int t = 0; t < T_; ++t) {
    const float* hrRe = hre[t & 1];
    const float* hrIm = him[t & 1];
    float* hwRe = hre[(t + 1) & 1];
    float* hwIm = him[(t + 1) & 1];

    Acc acc;
    cgemm_packed(hrRe, hrIm, H_, row0, base, base + PLANE_, base + 2 * PLANE_,
                 col0, H_, acc);

    if (gate < 2) {
#pragma unroll
      for (int q = 0; q < 8; ++q) {
        int m = q + 8 * half;
        int b = row0 + m;
        float2 px =
            *(const float2*)(xg + (((size_t)b * T_ + t) * H_ + col0 + n) * 2);
        rz[gate][m][n][0] = fast_sigmoid(acc.re[q] + px.x);
        rz[gate][m][n][1] = fast_sigmoid(acc.im[q] + px.y);
      }
    } else {
      float2 bb = *(const float2*)(bhn + (size_t)(col0 + n) * 2);
#pragma unroll
      for (int q = 0; q < 8; ++q) { acc.re[q] += bb.x; acc.im[q] += bb.y; }
    }
    __syncthreads();  // (A) r,z visible

    if (gate == 2) {
#pragma unroll
      for (int q = 0; q < 8; ++q) {
        int m = q + 8 * half;
        int b = row0 + m;
        size_t ho = (size_t)b * H_ + col0 + n;
        size_t xo = (((size_t)b * T_ + t) * H_ + col0 + n) * 2;
        float rr = rz[0][m][n][0], ri = rz[0][m][n][1];
        float ar = acc.re[q], ai = acc.im[q];   // h@Whn + bhn
        float tre = rr * ar - ri * ai;          // r * (...)
        float tim = rr * ai + ri * ar;
        float2 px = *(const float2*)(ys + xo);  // xn_t
        float nre = fast_tanh(px.x + tre);
        float nim = fast_tanh(px.y + tim);
        float zr = rz[1][m][n][0], zi = rz[1][m][n][1];
        float hpre = hrRe[ho], hpim = hrIm[ho];
        float ozr = 1.f - zr, ozi = -zi;        // (1 - z)
        float hnre = ozr * nre - ozi * nim + zr * hpre - zi * hpim;
        float hnim = ozr * nim + ozi * nre + zr * hpim + zi * hpre;
        hwRe[ho] = hnre;
        hwIm[ho] = hnim;
        *(float2*)(ys + xo) = make_float2(hnre, hnim);
        if (t == T_ - 1)
          *(float2*)(carry + ho * 2) = make_float2(hnre, hnim);
      }
    }
    __threadfence();
    __syncthreads();  // (B) h writes issued; LDS reuse safe
    if (threadIdx.x == 0) {
      __hip_atomic_fetch_add(ctr + btile, 1u, __ATOMIC_RELEASE,
                             __HIP_MEMORY_SCOPE_AGENT);
      unsigned target = 16u * (unsigned)(t + 1);
      while (__hip_atomic_load(ctr + btile, __ATOMIC_ACQUIRE,
                               __HIP_MEMORY_SCOPE_AGENT) < target) {
        __builtin_amdgcn_s_sleep(2);
      }
    }
    __syncthreads();  // all 16 WGs of this btile finished step t
  }
}

extern "C" void kernel_launch(void* const* d_in, const int* in_sizes, int n_in,
                              void* d_out, int out_size, void* d_ws,
                              size_t ws_size, hipStream_t stream) {
  (void)in_sizes; (void)n_in; (void)out_size; (void)ws_size;
  const float* c    = (const float*)d_in[0];
  const float* xs   = (const float*)d_in[1];
  const float* Wir  = (const float*)d_in[2];
  const float* Wiz  = (const float*)d_in[3];
  const float* Win  = (const float*)d_in[4];
  const float* bir  = (const float*)d_in[5];
  const float* biz  = (const float*)d_in[6];
  const float* bin_ = (const float*)d_in[7];
  const float* Whr  = (const float*)d_in[8];
  const float* Whz  = (const float*)d_in[9];
  const float* Whn  = (const float*)d_in[10];
  const float* bhn  = (const float*)d_in[11];

  float* out = (float*)d_out;
  float* carry = out;                     // [B,H] complex
  float* ys = out + (size_t)B_ * H_ * 2;  // [B,T,H] complex

  // ws layout (floats): [ctr x4 | pad 1KB][h planes x4][wpk][xr][xz]
  float* wsf = (float*)d_ws;
  unsigned* ctr = (unsigned*)d_ws;
  float* h0re = wsf + 256;
  float* h0im = h0re + (size_t)B_ * H_;
  float* h1re = h0im + (size_t)B_ * H_;
  float* h1im = h1re + (size_t)B_ * H_;
  float* wpk = h1im + (size_t)B_ * H_;              // 6*3*PLANE_ floats
  float* xr = wpk + (size_t)6 * 3 * PLANE_;
  float* xz = xr + (size_t)B_ * T_ * H_ * 2;

  hipMemsetAsync(ctr, 0, 4 * sizeof(unsigned), stream);

  // pack: 6*(F/2)*H weight sites + B*H c-split sites, 256 thr/block
  int packSites = 6 * (F_ / 2) * H_ + B_ * H_;
  cgru_pack_kernel<<<dim3((packSites + 255) / 256), 256, 0, stream>>>(
      Wir, Wiz, Win, Whr, Whz, Whn, c, wpk, h0re, h0im);

  cgru_proj_kernel<<<dim3(B_ * T_ / 16), 256, 0, stream>>>(
      xs, wpk, bir, biz, bin_, xr, xz, ys);

  cgru_scan_kernel<<<dim3(64), 96, 0, stream>>>(
      xr, xz, ys, wpk, bhn, h0re, h0im, h1re, h1im, ctr, carry);
}